// D_rho_81501299409146
// MI455X (gfx1250) — compile-verified
//
#include <hip/hip_runtime.h>

typedef float v2f __attribute__((ext_vector_type(2)));
typedef float v8f __attribute__((ext_vector_type(8)));

#define IMG 512
#define NPIX (IMG * IMG)
#define PP 36   // halo pitch (floats): 144B rows -> every (row, col%4==0) is 16B aligned
#define YP 20   // intermediate pitch: half-wave store groups hit disjoint banks

// --- fp32 WMMA 16x16x4: D = A(16x4) * B(4x16) + C ------------------------
static __device__ __forceinline__ v8f wmma4(v2f a, v2f b, v8f c) {
  // 8 args: (neg_a, A, neg_b, B, c_mod, C, reuse_a, reuse_b)
  return __builtin_amdgcn_wmma_f32_16x16x4_f32(false, a, false, b, (short)0, c,
                                               false, false);
}

// Band matrix for the 8-wide box window [p-3, p+4], K-tiles offset by -8:
// W[s][r] = 1 iff r+5 <= s <= r+12  -> single unsigned range test.
static __device__ __forceinline__ float band(int r, int s) {
  return ((unsigned)(s - r - 5) < 8u) ? 1.0f : 0.0f;
}

// Precompute the 8 per-lane band operands (identical for the horizontal
// B-operand and the vertical A-operand). The empty asm pins each value in a
// VGPR and makes it opaque, so the compiler cannot rematerialize the
// compare/select chain inside the WMMA loops.
static __device__ __forceinline__ void make_band(v2f* w, int lane) {
  const int half = lane >> 4;
  const int ln = lane & 15;
#pragma unroll
  for (int k = 0; k < 8; ++k) {
    const int s0 = k * 4 + 2 * half;
    float b0 = band(ln, s0);
    float b1 = band(ln, s0 + 1);
    asm("" : "+v"(b0), "+v"(b1));  // keep in VGPRs; computed exactly once
    w[k] = (v2f){b0, b1};
  }
}

// Box-sum of a 16x16 tile from a 32x32 LDS halo (origin tile_origin-8).
// Horizontal: Y(32x16) = P(32x32) * W(32x16); Vertical: D = V(16x32) * Y.
// MODE 0: integrand pa; MODE 1: pa*pa; MODE 2: pa*pb (fused at A-fetch).
// Lane l holds D rows {g + 8*(l>=16)}, col l&15 (standard C/D layout).
template <int MODE>
static __device__ __forceinline__ v8f boxsum32(const float* __restrict__ pa,
                                               const float* __restrict__ pb,
                                               float* __restrict__ y,
                                               const v2f* __restrict__ w,
                                               int lane) {
  const int half = lane >> 4;
  const int ln = lane & 15;
#pragma unroll
  for (int rb = 0; rb < 2; ++rb) {
    v8f acc = {0.f, 0.f, 0.f, 0.f, 0.f, 0.f, 0.f, 0.f};
    const int row = rb * 16 + ln;  // A-matrix M = lane&15 for all lanes
#pragma unroll
    for (int k = 0; k < 8; ++k) {
      const int c0 = k * 4 + 2 * half;  // A K index: lanes>=16 carry K+2
      float a0 = pa[row * PP + c0];
      float a1 = pa[row * PP + c0 + 1];
      if (MODE == 1) { a0 *= a0; a1 *= a1; }
      if (MODE == 2) {
        a0 *= pb[row * PP + c0];
        a1 *= pb[row * PP + c0 + 1];
      }
      acc = wmma4((v2f){a0, a1}, w[k], acc);
    }
#pragma unroll
    for (int g = 0; g < 8; ++g)
      y[(rb * 16 + g + 8 * half) * YP + ln] = acc[g];
  }
  __syncthreads();  // y fully written (cross-lane) before vertical pass
  v8f acc = {0.f, 0.f, 0.f, 0.f, 0.f, 0.f, 0.f, 0.f};
#pragma unroll
  for (int k = 0; k < 8; ++k) {
    const int t0 = k * 4 + 2 * half;
    v2f b = {y[t0 * YP + ln], y[(t0 + 1) * YP + ln]};  // Y as B-operand
    acc = wmma4(w[k], b, acc);
  }
  return acc;
}

// Border tiles: predicated scalar staging with zero padding.
static __device__ __forceinline__ void load_halo_scalar(
    const float* __restrict__ src, int I, int J, int lane,
    float* __restrict__ p) {
#pragma unroll 4
  for (int t = 0; t < 32; ++t) {
    const int idx = t * 32 + lane;   // consecutive lanes -> consecutive cols
    const int r = idx >> 5, c = idx & 31;
    const int gr = I - 8 + r, gc = J - 8 + c;
    float v = 0.0f;
    if ((unsigned)gr < IMG && (unsigned)gc < IMG) v = src[gr * IMG + gc];
    p[r * PP + c] = v;
  }
}

// Interior tiles: stage the 32x32 halo straight into LDS with async b128
// transfers (no VGPR round trip, tracked by ASYNCcnt). 8 instructions x
// (32 lanes x 16B) = 4KB. GVS mode: addr = SADDR(64b) + VADDR(32b offset).
static __device__ __forceinline__ void load_halo_async(
    const float* __restrict__ src, int I, int J, int lane,
    float* __restrict__ p) {
  const unsigned ldsbase = (unsigned)(unsigned long long)(const void*)p;
  const int r0 = lane >> 3;        // 0..3
  const int c0 = (lane & 7) * 4;   // 0,4,...,28
#pragma unroll
  for (int t = 0; t < 8; ++t) {
    const int r = t * 4 + r0;
    const unsigned lds = ldsbase + (unsigned)((r * PP + c0) * 4);
    const int goff = ((I - 8 + r) * IMG + (J - 8 + c0)) * 4;
    asm volatile("global_load_async_to_lds_b128 %0, %1, %2"
                 :
                 : "v"(lds), "v"(goff), "s"(src)
                 : "memory");
  }
  asm volatile("s_wait_asynccnt 0" ::: "memory");
}

static __device__ __forceinline__ void load_halo(const float* __restrict__ src,
                                                 int I, int J, int lane,
                                                 float* __restrict__ p,
                                                 bool interior) {
  if (interior) load_halo_async(src, I, J, lane, p);
  else          load_halo_scalar(src, I, J, lane, p);
}

// Before overwriting a staging buffer that earlier LDS reads used, drain DS.
static __device__ __forceinline__ void wait_lds_reads() {
  asm volatile("s_wait_dscnt 0" ::: "memory");
}

// ---- Stage 1a: centered images  c = x - box(x)/64  (40 images) -----------
__global__ __launch_bounds__(32) void k_center(const float* __restrict__ outs,
                                               const float* __restrict__ labs,
                                               float* __restrict__ ac,
                                               float* __restrict__ bc) {
  __shared__ __align__(16) float p[32 * PP];
  __shared__ float y[32 * YP];
  const int img = blockIdx.y;
  const float* src;
  float* dst;
  if (img < 24) { src = outs + (size_t)img * NPIX; dst = ac + (size_t)img * NPIX; }
  else          { src = labs + (size_t)(img - 24) * NPIX; dst = bc + (size_t)(img - 24) * NPIX; }
  const int tile = blockIdx.x;
  const int I = (tile >> 5) * 16, J = (tile & 31) * 16;
  const int lane = threadIdx.x, half = lane >> 4, ln = lane & 15;
  const bool interior = (I >= 16 && I <= 480 && J >= 16 && J <= 480);

  v2f w[8];
  make_band(w, lane);
  load_halo(src, I, J, lane, p, interior);
  __syncthreads();
  v8f s = boxsum32<0>(p, p, y, w, lane);
#pragma unroll
  for (int g = 0; g < 8; ++g) {
    const int rr = g + 8 * half;
    const float mu = s[g] * (1.0f / 64.0f);
    dst[(size_t)(I + rr) * IMG + (J + ln)] = p[(8 + rr) * PP + (8 + ln)] - mu;
  }
}

// ---- Stage 1b: sii = max(box(c*c), 1e-20)  (40 images) -------------------
__global__ __launch_bounds__(32) void k_var(const float* __restrict__ ac,
                                            const float* __restrict__ bc,
                                            float* __restrict__ sii,
                                            float* __restrict__ sjj) {
  __shared__ __align__(16) float p[32 * PP];
  __shared__ float y[32 * YP];
  const int img = blockIdx.y;
  const float* src;
  float* dst;
  if (img < 24) { src = ac + (size_t)img * NPIX; dst = sii + (size_t)img * NPIX; }
  else          { src = bc + (size_t)(img - 24) * NPIX; dst = sjj + (size_t)(img - 24) * NPIX; }
  const int tile = blockIdx.x;
  const int I = (tile >> 5) * 16, J = (tile & 31) * 16;
  const int lane = threadIdx.x, half = lane >> 4, ln = lane & 15;
  const bool interior = (I >= 16 && I <= 480 && J >= 16 && J <= 480);

  v2f w[8];
  make_band(w, lane);
  load_halo(src, I, J, lane, p, interior);
  __syncthreads();
  v8f s = boxsum32<1>(p, p, y, w, lane);  // square fused at A-fetch
#pragma unroll
  for (int g = 0; g < 8; ++g) {
    const int rr = g + 8 * half;
    dst[(size_t)(I + rr) * IMG + (J + ln)] = fmaxf(s[g], 1e-20f);
  }
}

// ---- Stage 2: sij = box(ac*bc); cc; max over j; accumulate means ---------
__global__ __launch_bounds__(32) void k_pair(const float* __restrict__ ac,
                                             const float* __restrict__ bc,
                                             const float* __restrict__ sii,
                                             const float* __restrict__ sjj,
                                             double* __restrict__ accum) {
  __shared__ __align__(16) float pa[32 * PP];
  __shared__ __align__(16) float pb[32 * PP];
  __shared__ float y[32 * YP];
  const int bi = blockIdx.y;  // b*6 + i
  const int b = bi / 6;
  const int tile = blockIdx.x;
  const int I = (tile >> 5) * 16, J = (tile & 31) * 16;
  const int lane = threadIdx.x, half = lane >> 4, ln = lane & 15;
  const bool interior = (I >= 16 && I <= 480 && J >= 16 && J <= 480);

  v2f w[8];
  make_band(w, lane);

  const float* A = ac + (size_t)bi * NPIX;
  const float* SA = sii + (size_t)bi * NPIX;
  load_halo(A, I, J, lane, pa, interior);

  float siv[8], ymax[8];
#pragma unroll
  for (int g = 0; g < 8; ++g) {
    siv[g] = SA[(size_t)(I + g + 8 * half) * IMG + (J + ln)];
    ymax[g] = -1.0f;  // Y = max(cc, -1) folded into the j-max init
  }

  for (int j = 0; j < 4; ++j) {
    const float* B = bc + (size_t)(b * 4 + j) * NPIX;
    const float* SB = sjj + (size_t)(b * 4 + j) * NPIX;
    wait_lds_reads();   // prior iter's pb/y reads done before overwrite
    __syncthreads();
    load_halo(B, I, J, lane, pb, interior);
    __syncthreads();
    v8f sij = boxsum32<2>(pa, pb, y, w, lane);
#pragma unroll
    for (int g = 0; g < 8; ++g) {
      const float sjv = SB[(size_t)(I + g + 8 * half) * IMG + (J + ln)];
      const float denom = sqrtf(siv[g] * sjv) + 1e-20f;
      ymax[g] = fmaxf(ymax[g], sij[g] / denom);
    }
  }

  float part = 0.0f;
#pragma unroll
  for (int g = 0; g < 8; ++g) part += 1.0f - fminf(ymax[g], 1.0f);
  // wave32 reduction
  for (int off = 16; off; off >>= 1) part += __shfl_xor(part, off, 32);
  if (lane == 0) atomicAdd(&accum[bi], (double)part);
}

__global__ __launch_bounds__(32) void k_zero(double* __restrict__ accum) {
  if (threadIdx.x < 24) accum[threadIdx.x] = 0.0;
}

__global__ __launch_bounds__(32) void k_final(const double* __restrict__ accum,
                                              float* __restrict__ out) {
  const int t = threadIdx.x;
  if (t < 24) out[1 + t] = (float)(accum[t] / (double)NPIX);
  if (t == 0) {
    double s = 0.0;
    for (int k = 0; k < 24; ++k) s += accum[k];
    out[0] = (float)(s / ((double)NPIX * 24.0));
  }
}

extern "C" void kernel_launch(void* const* d_in, const int* in_sizes, int n_in,
                              void* d_out, int out_size, void* d_ws,
                              size_t ws_size, hipStream_t stream) {
  (void)in_sizes; (void)n_in; (void)out_size; (void)ws_size;
  const float* outs = (const float*)d_in[0];  // (4,6,512,512)
  const float* labs = (const float*)d_in[1];  // (4,4,512,512)
  float* ws = (float*)d_ws;
  float* ac  = ws;                          // 24 * NPIX
  float* sii = ac + (size_t)24 * NPIX;      // 24 * NPIX
  float* bc  = sii + (size_t)24 * NPIX;     // 16 * NPIX
  float* sjj = bc + (size_t)16 * NPIX;      // 16 * NPIX
  double* accum = (double*)(sjj + (size_t)16 * NPIX);  // 24 doubles (8B-aligned)

  const dim3 blk(32, 1, 1);
  k_zero<<<dim3(1), blk, 0, stream>>>(accum);
  k_center<<<dim3(1024, 40), blk, 0, stream>>>(outs, labs, ac, bc);
  k_var<<<dim3(1024, 40), blk, 0, stream>>>(ac, bc, sii, sjj);
  k_pair<<<dim3(1024, 24), blk, 0, stream>>>(ac, bc, sii, sjj, accum);
  k_final<<<dim3(1), blk, 0, stream>>>(accum, (float*)d_out);
}